// MLP_13451837571858
// MI455X (gfx1250) — compile-verified
//
#include <hip/hip_runtime.h>
#include <math.h>

typedef __attribute__((ext_vector_type(2))) float v2f;
typedef __attribute__((ext_vector_type(8))) float v8f;
typedef int vi4 __attribute__((vector_size(4 * sizeof(int))));   // matches builtin param type

#define B_SZ  1024
#define D_IN  784
#define HID   128
#define D_OUT 10

#if defined(__has_builtin)
#if __has_builtin(__builtin_amdgcn_global_load_async_to_lds_b128)
#define HAVE_ASYNC_LDS 1
#endif
#endif
#ifndef HAVE_ASYNC_LDS
#define HAVE_ASYNC_LDS 0
#endif

// Signed power-of-two part: mantissa bits cleared. P(0) = +/-0.
__device__ __forceinline__ float p2(float v) {
    return __uint_as_float(__float_as_uint(v) & 0xFF800000u);
}

// Fused L-Mul MLP: relu(lmul_linear(x,W1,b1)) -> lmul_linear(.,W2,b2) -> log_softmax
// One block = one 16-row tile of the batch. 8 waves (wave32) cover all 128 hidden cols.
__global__ __launch_bounds__(256) void MLP_lmul_wmma_kernel(
    const float* __restrict__ X,  const float* __restrict__ W1,
    const float* __restrict__ b1, const float* __restrict__ W2,
    const float* __restrict__ b2, float* __restrict__ out)
{
    __shared__ float sX[16][D_IN + 4];   // 16 x 784 input tile (+4 pad: stride 788, conflict-free)
    __shared__ float sH[16][HID + 4];    // 16 x 128 hidden tile (+pad)

    const int tid  = threadIdx.x;
    const int wave = tid >> 5;
    const int lane = tid & 31;
    const int half = lane >> 4;          // 0: K=0..1 / M rows 0..7 ; 1: K=2..3 / M rows 8..15
    const int r    = lane & 15;
    const int m0   = blockIdx.x * 16;    // batch-row tile base

    // ---- stage X tile into LDS: 16 rows x 784 floats = 3136 float4 chunks ----
    // Async global->LDS (ASYNCcnt-tracked, no VGPR bounce) when available.
    for (int idx = tid; idx < 16 * 196; idx += 256) {
        const int row = idx / 196;
        const int col = (idx - row * 196) * 4;                 // 16B-aligned both sides
        const float* g = X + (size_t)(m0 + row) * D_IN + col;
        float*       l = &sX[row][col];
#if HAVE_ASYNC_LDS
        __builtin_amdgcn_global_load_async_to_lds_b128(
            (__attribute__((address_space(1))) vi4*)g,
            (__attribute__((address_space(3))) vi4*)l, 0, 0);
#else
        *(float4*)l = *(const float4*)g;
#endif
    }
#if HAVE_ASYNC_LDS
    asm volatile("s_wait_asynccnt 0x0" ::: "memory");          // drain this wave's async copies
#endif
    __syncthreads();                                           // all waves' copies complete

    // ================= Layer 1: wave handles output cols [n0, n0+16) =================
    const int n0 = wave * 16;
    const float* wrow = W1 + (size_t)(n0 + r) * D_IN;          // lane's B-row (N = n0+r)

    v8f c = {0.f,0.f,0.f,0.f,0.f,0.f,0.f,0.f};

    // Doubled-K exact-GEMM identity:
    //   sum_i lmul(x_i,w_i) = sum_i (2x_i - 1.875*P(x_i))*P(w_i) + sum_i 2*P(x_i)*w_i
    // Pass 1: A = 2x - 1.875*P(x), B = P(w)
    #pragma unroll 4
    for (int k = 0; k < D_IN; k += 4) {
        const int i = k + 2 * half;
        float x0 = sX[r][i], x1 = sX[r][i + 1];
        float w0 = wrow[i],  w1 = wrow[i + 1];
        v2f a = { 2.0f * x0 - 1.875f * p2(x0), 2.0f * x1 - 1.875f * p2(x1) };
        v2f b = { p2(w0), p2(w1) };
        c = __builtin_amdgcn_wmma_f32_16x16x4_f32(false, a, false, b, (short)0, c, false, false);
    }
    // Pass 2: A = 2*P(x), B = w
    #pragma unroll 4
    for (int k = 0; k < D_IN; k += 4) {
        const int i = k + 2 * half;
        float x0 = sX[r][i], x1 = sX[r][i + 1];
        float w0 = wrow[i],  w1 = wrow[i + 1];
        v2f a = { 2.0f * p2(x0), 2.0f * p2(x1) };
        v2f b = { w0, w1 };
        c = __builtin_amdgcn_wmma_f32_16x16x4_f32(false, a, false, b, (short)0, c, false, false);
    }

    // bias + relu -> LDS hidden tile. C layout: VGPR j holds (M = j + 8*half, N = lane&15).
    {
        const float bias = b1[n0 + r];
        #pragma unroll
        for (int j = 0; j < 8; ++j) {
            float h = c[j] + bias;
            sH[j + 8 * half][n0 + r] = fmaxf(h, 0.0f);
        }
    }
    __syncthreads();

    // ================= Layer 2 + log_softmax on wave 0 (N=10 padded to 16) =================
    if (wave == 0) {
        const bool valid = (r < D_OUT);
        const float* w2row = W2 + (size_t)r * HID;             // only meaningful when valid

        v8f c2 = {0.f,0.f,0.f,0.f,0.f,0.f,0.f,0.f};

        #pragma unroll 4
        for (int k = 0; k < HID; k += 4) {
            const int i = k + 2 * half;
            float h0 = sH[r][i], h1 = sH[r][i + 1];
            float w0 = valid ? w2row[i]     : 0.0f;            // zero cols: P(0)=0 -> exact padding
            float w1 = valid ? w2row[i + 1] : 0.0f;
            v2f a = { 2.0f * h0 - 1.875f * p2(h0), 2.0f * h1 - 1.875f * p2(h1) };
            v2f b = { p2(w0), p2(w1) };
            c2 = __builtin_amdgcn_wmma_f32_16x16x4_f32(false, a, false, b, (short)0, c2, false, false);
        }
        #pragma unroll 4
        for (int k = 0; k < HID; k += 4) {
            const int i = k + 2 * half;
            float h0 = sH[r][i], h1 = sH[r][i + 1];
            float w0 = valid ? w2row[i]     : 0.0f;
            float w1 = valid ? w2row[i + 1] : 0.0f;
            v2f a = { 2.0f * p2(h0), 2.0f * p2(h1) };
            v2f b = { w0, w1 };
            c2 = __builtin_amdgcn_wmma_f32_16x16x4_f32(false, a, false, b, (short)0, c2, false, false);
        }

        const float bias2 = valid ? b2[r] : 0.0f;

        #pragma unroll
        for (int j = 0; j < 8; ++j) {
            float v  = c2[j] + bias2;                          // logit (row m0+j+8*half, col r)
            float vm = valid ? v : -INFINITY;
            vm = fmaxf(vm, __shfl_xor(vm, 8, 16));
            vm = fmaxf(vm, __shfl_xor(vm, 4, 16));
            vm = fmaxf(vm, __shfl_xor(vm, 2, 16));
            vm = fmaxf(vm, __shfl_xor(vm, 1, 16));
            float ex = valid ? __expf(v - vm) : 0.0f;
            float s  = ex;
            s += __shfl_xor(s, 8, 16);
            s += __shfl_xor(s, 4, 16);
            s += __shfl_xor(s, 2, 16);
            s += __shfl_xor(s, 1, 16);
            if (valid) {
                const int row = m0 + j + 8 * half;
                out[(size_t)row * D_OUT + r] = (v - vm) - __logf(s);
            }
        }
    }
}

extern "C" void kernel_launch(void* const* d_in, const int* in_sizes, int n_in,
                              void* d_out, int out_size, void* d_ws, size_t ws_size,
                              hipStream_t stream) {
    (void)in_sizes; (void)n_in; (void)d_ws; (void)ws_size; (void)out_size;
    const float* X  = (const float*)d_in[0];
    const float* W1 = (const float*)d_in[1];
    const float* b1 = (const float*)d_in[2];
    const float* W2 = (const float*)d_in[3];
    const float* b2 = (const float*)d_in[4];
    float* out = (float*)d_out;

    dim3 grid(B_SZ / 16);   // 64 blocks, one per 16-row batch tile
    dim3 block(256);        // 8 wave32 waves
    MLP_lmul_wmma_kernel<<<grid, block, 0, stream>>>(X, W1, b1, W2, b2, out);
}